// HieraLossMapillary_23673859736481
// MI455X (gfx1250) — compile-verified
//
#include <hip/hip_runtime.h>
#include <stdint.h>

// Problem constants (fixed by the reference)
#define HWPIX   (512*512)        // pixels per image plane = 262144
#define NCH     148              // 124 + 17 + 7 prediction channels
#define NTHR1   256
#define NBLK1   512              // 131072 threads, 4 pixels (float4) each
#define NITER   148              // per-thread channel iterations (2 batches x 148 ch / step 2)
#define STEPB   (2u*HWPIX*4u)    // byte stride per iteration = 2 MiB

typedef __attribute__((ext_vector_type(2))) float v2f;
typedef __attribute__((ext_vector_type(8))) float v8f;

// ---------------- async copy helpers (CDNA5 ASYNCcnt path) ----------------
__device__ __forceinline__ void async_ld_b128(uint32_t lds_addr, uint32_t voff,
                                              unsigned long long sbase) {
  // GVS mode: mem = SADDR(64) + VADDR(u32). LDS dest byte addr in VDST vgpr.
  asm volatile("global_load_async_to_lds_b128 %0, %1, %2"
               :: "v"(lds_addr), "v"(voff), "s"(sbase)
               : "memory");
}

#if __has_builtin(__builtin_amdgcn_s_wait_asynccnt)
#define WAIT_ASYNC(n) __builtin_amdgcn_s_wait_asynccnt(n)
#else
#define WAIT_ASYNC(n) asm volatile("s_wait_asynccnt %0" :: "i"(n) : "memory")
#endif

// ---------------- hierarchy remap (replaces 256-entry LUTs) ----------------
__device__ __forceinline__ int mid_idx(int lab) {
  const int b[16] = {2,12,26,30,32,35,38,55,59,66,78,84,90,96,105,117};
  int m = 0;
#pragma unroll
  for (int i = 0; i < 16; ++i) m += (lab >= b[i]) ? 1 : 0;
  return m;   // [117,123) -> 16, and 255 -> 16 (matches LUT default)
}
__device__ __forceinline__ int high_idx(int lab) {
  const int b[6] = {2,30,35,59,66,117};
  int m = 0;
#pragma unroll
  for (int i = 0; i < 6; ++i) m += (lab >= b[i]) ? 1 : 0;
  return m;   // default 6
}

// ---------------- main streaming kernel ----------------
__global__ __launch_bounds__(NTHR1)
void hiera_asl_main(const float* __restrict__ cls,
                    const int* __restrict__ label,
                    float* __restrict__ part_loss,
                    float* __restrict__ part_valid) {
  __shared__ float4 stage[2][NTHR1];   // async double buffer (8 KB)
  __shared__ float  red[NTHR1];
  __shared__ float  redv[NTHR1];

  const int tid = threadIdx.x;
  const int t   = blockIdx.x * NTHR1 + tid;       // 0..131071
  const uint32_t e0   = 4u * (uint32_t)t;         // first element index
  const int      pix  = (int)(e0 & (HWPIX - 1));  // pixel column block (invariant)
  const int      t_hi = (int)(e0 >> 18);          // 0 or 1 (starting channel parity)

  // Per-pixel packed targets: tgt[7:0] | mid[15:8] | high[23:16] | valid[24]
  uint32_t pk0[4], pk1[4];
  int vcount = 0;
#pragma unroll
  for (int k = 0; k < 4; ++k) {
    int l0 = label[pix + k];
    int l1 = label[HWPIX + pix + k];
    int tg0 = (l0 == 255) ? 123 : l0;
    int tg1 = (l1 == 255) ? 123 : l1;
    int v0 = (l0 != 255) ? 1 : 0;
    int v1 = (l1 != 255) ? 1 : 0;
    pk0[k] = (uint32_t)tg0 | ((uint32_t)mid_idx(l0) << 8) |
             ((uint32_t)high_idx(l0) << 16) | ((uint32_t)v0 << 24);
    pk1[k] = (uint32_t)tg1 | ((uint32_t)mid_idx(l1) << 8) |
             ((uint32_t)high_idx(l1) << 16) | ((uint32_t)v1 << 24);
    vcount += t_hi ? v1 : v0;   // each (b,pixel) counted by exactly one thread
  }

  const unsigned long long sbase = (unsigned long long)(uintptr_t)cls;
  const uint32_t laddr0 = (uint32_t)(uintptr_t)&stage[0][tid];
  const uint32_t laddr1 = (uint32_t)(uintptr_t)&stage[1][tid];

  uint32_t voff = e0 * 4u;                 // byte offset, max ~310 MB < 2^31
  async_ld_b128(laddr0, voff, sbase);      // prologue: stage iteration 0
  voff += STEPB;

  float acc = 0.0f;
  for (int i = 0; i < NITER; ++i) {
    if (i + 1 < NITER) {
      async_ld_b128((((i + 1) & 1) ? laddr1 : laddr0), voff, sbase);
      voff += STEPB;
      WAIT_ASYNC(1);                       // older load (iter i) complete
    } else {
      WAIT_ASYNC(0);
    }
    float4 x4 = stage[i & 1][tid];

    // channel bookkeeping (wave-uniform)
    const int g  = t_hi + 2 * i;           // global channel index 0..295
    const int bB = (g >= NCH) ? 1 : 0;     // batch select
    const int c  = g - (bB ? NCH : 0);
    uint32_t shift; int cc;
    if (c < 124)      { shift = 0;  cc = c; }
    else if (c < 141) { shift = 8;  cc = c - 124; }
    else              { shift = 16; cc = c - 141; }

    const float xs[4] = {x4.x, x4.y, x4.z, x4.w};
#pragma unroll
    for (int k = 0; k < 4; ++k) {
      const uint32_t info = bB ? pk1[k] : pk0[k];
      const bool  pos  = (((info >> shift) & 0xffu) == (uint32_t)cc);
      const float vfl  = (float)(info >> 24);
      const float x    = xs[k];
      const float ex   = __expf(-x);                         // v_exp_f32
      const float p    = __builtin_amdgcn_rcpf(1.0f + ex);   // v_rcp_f32
      const float onep = 1.0f - p;
      const float q    = fminf(onep + 0.05f, 1.0f);          // xs_neg, >= 0.05
      const float dn   = fmaxf(p - 0.05f, 0.0f);             // 1 - xs_neg
      const float dn2  = dn * dn;
      const float arg  = pos ? fmaxf(p, 1e-8f) : q;
      const float w    = pos ? onep : dn2 * dn2;             // (1-p)^1 / (1-q)^4
      acc = fmaf(__logf(arg) * w, vfl, acc);                 // v_log_f32
    }
  }

  // deterministic block tree reduction
  __syncthreads();
  red[tid]  = acc;
  redv[tid] = (float)vcount;
  __syncthreads();
#pragma unroll
  for (int s = NTHR1 / 2; s > 0; s >>= 1) {
    if (tid < s) { red[tid] += red[tid + s]; redv[tid] += redv[tid + s]; }
    __syncthreads();
  }
  if (tid == 0) {
    part_loss[blockIdx.x]  = red[0];
    part_valid[blockIdx.x] = redv[0];
  }
}

// ---------------- final reduction: 512 partials via WMMA ones-trick ----------------
#if __has_builtin(__builtin_amdgcn_wmma_f32_16x16x4_f32)
__global__ void hiera_asl_final(const float* __restrict__ pl,
                                const float* __restrict__ pv,
                                float* __restrict__ out) {
  const int lane = threadIdx.x;            // 32 threads = 1 full wave (EXEC all 1s)
  v2f ones; ones.x = 1.0f; ones.y = 1.0f;  // A = ones(16x4) -> D[m,n] = sum_k B[k,n] + C
  v8f cl = {}; v8f cv = {};
#pragma unroll
  for (int j = 0; j < 8; ++j) {            // 8 chunks x 64 values = 512 partials
    const int base = j * 64 + lane * 2;
    v2f bl; bl.x = pl[base]; bl.y = pl[base + 1];
    v2f bv; bv.x = pv[base]; bv.y = pv[base + 1];
    cl = __builtin_amdgcn_wmma_f32_16x16x4_f32(false, ones, false, bl, (short)0, cl, false, false);
    cv = __builtin_amdgcn_wmma_f32_16x16x4_f32(false, ones, false, bv, (short)0, cv, false, false);
  }
  // D row 0 = lanes 0..15 of first C/D VGPR; sum over n gives the exact total
  float sl = cl[0];
  float sv = cv[0];
#pragma unroll
  for (int m = 1; m <= 8; m <<= 1) {
    sl += __shfl_xor(sl, m, 32);
    sv += __shfl_xor(sv, m, 32);
  }
  if (lane == 0) out[0] = -sl / fmaxf(sv, 1.0f);
}
#else
__global__ void hiera_asl_final(const float* __restrict__ pl,
                                const float* __restrict__ pv,
                                float* __restrict__ out) {
  const int lane = threadIdx.x;
  float sl = 0.0f, sv = 0.0f;
  for (int i = lane; i < NBLK1; i += 32) { sl += pl[i]; sv += pv[i]; }
#pragma unroll
  for (int m = 1; m <= 16; m <<= 1) {
    sl += __shfl_xor(sl, m, 32);
    sv += __shfl_xor(sv, m, 32);
  }
  if (lane == 0) out[0] = -sl / fmaxf(sv, 1.0f);
}
#endif

extern "C" void kernel_launch(void* const* d_in, const int* in_sizes, int n_in,
                              void* d_out, int out_size, void* d_ws, size_t ws_size,
                              hipStream_t stream) {
  (void)in_sizes; (void)n_in; (void)out_size; (void)ws_size;
  const float* cls   = (const float*)d_in[0];
  const int*   label = (const int*)d_in[1];
  float* out = (float*)d_out;
  float* pl  = (float*)d_ws;          // 512 loss partials
  float* pv  = pl + NBLK1;            // 512 valid-count partials

  hiera_asl_main<<<NBLK1, NTHR1, 0, stream>>>(cls, label, pl, pv);
  hiera_asl_final<<<1, 32, 0, stream>>>(pl, pv, out);
}